// CRF_53283364274652
// MI455X (gfx1250) — compile-verified
//
#include <hip/hip_runtime.h>
#include <hip/hip_bf16.h>

#define BATCH 1024
#define TLEN  512
#define LDIM  32

typedef float v2f __attribute__((ext_vector_type(2)));
typedef float v8f __attribute__((ext_vector_type(8)));

__device__ __forceinline__ float fast_exp2(float x) { return __builtin_amdgcn_exp2f(x); }
__device__ __forceinline__ float fast_log2(float x) { return __builtin_amdgcn_logf(x); }

// ---------------------------------------------------------------------------
// Forward (sum-over-paths) scan.  One wave per block; each wave owns 16
// batches.  State Z kept as (offset c, bounded z) with z in WMMA C/D layout:
//   tile0 = states 0..15, tile1 = states 16..31
//   VGPR r: lanes 0-15 -> batch r, lanes 16-31 -> batch r+8
// Step:  m = z[batch, state0] (broadcast; any per-batch offset is exact since
//        it is added back through c — only needs to bound the exp range);
//        E = exp2((z-m)/ln2);  S = E @ exp(trans) via v_wmma_f32_16x16x4_f32
//        (8 K-steps x 2 N-halves, split into even/odd partial chains);
//        z' = log2(S)*ln2 + h_t;  c += m.
// Final: sum_over_path[b] = 32*c[b] + sum_s z[b][s].
// ---------------------------------------------------------------------------
__global__ __launch_bounds__(32) void crf_scan_kernel(
    const float* __restrict__ inputs,   // (B, T, L)
    const float* __restrict__ trans,    // (L, L)
    float* __restrict__ sumpath)        // (B)
{
    constexpr float RCP_LN2 = 1.44269504088896340736f;
    constexpr float LN2     = 0.69314718055994530942f;
    constexpr int   PITCH   = 36;       // LDS row pitch (floats): bank-conflict-free

    __shared__ float esh[16 * PITCH];   // 16x32 E tile (padded)

    const int lane = threadIdx.x & 31;
    const int hi   = lane >> 4;         // lane-half: 0 or 1
    const int st   = lane & 15;         // column-within-group
    const int wb   = blockIdx.x * 16;   // first batch of this wave
    const int bsrc = hi << 4;           // broadcast source lane (0 or 16)

    // --- Precompute B operands: Wp[l][m] = exp2(trans[l][m]/ln2) = exp(trans)
    // B layout (4x16 f32): VGPR0 lanes0-15 = K=4k row, lanes16-31 = K=4k+2 row;
    // VGPR1 = next K row.  Per lane: row = 4k+2*hi (+1 for .y), col = 16*half+st.
    v2f bw[8][2];
    #pragma unroll
    for (int k = 0; k < 8; ++k) {
        const int row = 4 * k + 2 * hi;
        #pragma unroll
        for (int hf = 0; hf < 2; ++hf) {
            const int col = 16 * hf + st;
            bw[k][hf].x = fast_exp2(trans[(row    ) * LDIM + col] * RCP_LN2);
            bw[k][hf].y = fast_exp2(trans[(row + 1) * LDIM + col] * RCP_LN2);
        }
    }

    // h-load base in D layout: element (r, tile) = inputs[wb+r+8*hi][t][st+16*tile]
    const float* hbase = inputs + (size_t)(wb + 8 * hi) * TLEN * LDIM + st;

    // --- Init: Z0 = inputs[:, 0, :]
    float z0[8], z1[8], coff[8];
    #pragma unroll
    for (int r = 0; r < 8; ++r) {
        const float* p = hbase + (size_t)r * TLEN * LDIM;
        z0[r]   = p[0];
        z1[r]   = p[16];
        coff[r] = 0.0f;
    }

    const int ldsLoadBase  = PITCH * st + 2 * hi;         // A-operand load base (floats)
    const int ldsStoreBase = PITCH * 8 * hi + st;         // E store base (row r+8*hi)

    for (int t = 1; t < TLEN; ++t) {
        // 1) per-batch offset: broadcast z[batch, state 0] across the row's lanes
        float mx[8];
        #pragma unroll
        for (int r = 0; r < 8; ++r)
            mx[r] = __shfl(z0[r], bsrc, 32);

        // 2) E = exp2((z - m) / ln2), stored to LDS in batch-row-major form
        #pragma unroll
        for (int r = 0; r < 8; ++r) {
            const float e0 = fast_exp2((z0[r] - mx[r]) * RCP_LN2);
            const float e1 = fast_exp2((z1[r] - mx[r]) * RCP_LN2);
            esh[ldsStoreBase + r * PITCH]      = e0;
            esh[ldsStoreBase + r * PITCH + 16] = e1;
        }
        // (same-wave DS ops are in-order; register deps handle the rest)

        // 3) S = E @ Wp : 8 K-steps, A operand re-read in 16x4 f32 A-layout.
        //    Even/odd K split -> 4 independent WMMA accumulation chains.
        v8f acc0a = {}, acc0b = {};
        v8f acc1a = {}, acc1b = {};
        #pragma unroll
        for (int k = 0; k < 8; k += 2) {
            const v2f ae = *(const v2f*)&esh[ldsLoadBase + 4 * k];
            const v2f ao = *(const v2f*)&esh[ldsLoadBase + 4 * (k + 1)];
            acc0a = __builtin_amdgcn_wmma_f32_16x16x4_f32(
                        false, ae, false, bw[k][0],     (short)0, acc0a, false, false);
            acc1a = __builtin_amdgcn_wmma_f32_16x16x4_f32(
                        false, ae, false, bw[k][1],     (short)0, acc1a, false, false);
            acc0b = __builtin_amdgcn_wmma_f32_16x16x4_f32(
                        false, ao, false, bw[k + 1][0], (short)0, acc0b, false, false);
            acc1b = __builtin_amdgcn_wmma_f32_16x16x4_f32(
                        false, ao, false, bw[k + 1][1], (short)0, acc1b, false, false);
        }
        const v8f acc0 = acc0a + acc0b;
        const v8f acc1 = acc1a + acc1b;

        // 4) z' = ln(S) + h_t ;  c += m
        const float* hp = hbase + (size_t)t * LDIM;
        #pragma unroll
        for (int r = 0; r < 8; ++r) {
            const float* p = hp + (size_t)r * TLEN * LDIM;
            z0[r] = fast_log2(acc0[r]) * LN2 + p[0];
            z1[r] = fast_log2(acc1[r]) * LN2 + p[16];
            coff[r] += mx[r];
        }
    }

    // --- Final: sum over 32 states + 32 * offset
    #pragma unroll
    for (int r = 0; r < 8; ++r) {
        float s = z0[r] + z1[r];
        s += __shfl_xor(s, 1, 32);
        s += __shfl_xor(s, 2, 32);
        s += __shfl_xor(s, 4, 32);
        s += __shfl_xor(s, 8, 32);
        if (st == 0)
            sumpath[wb + r + 8 * hi] = s + 32.0f * coff[r];
    }
}

// ---------------------------------------------------------------------------
// Path score.  One block per batch.  Labels are exact one-hot: recover the
// index as sum(l * label[l]); emission via exact dot; transition gathered
// from LDS-resident trans.
// ---------------------------------------------------------------------------
__global__ __launch_bounds__(256) void crf_path_kernel(
    const float* __restrict__ inputs,
    const float* __restrict__ labels,
    const float* __restrict__ trans,
    float* __restrict__ path)           // (B)
{
    __shared__ float trs[LDIM * LDIM];
    __shared__ int   idxs[TLEN];
    __shared__ float red[256];

    const int b   = blockIdx.x;
    const int tid = threadIdx.x;

    for (int i = tid; i < LDIM * LDIM; i += 256) trs[i] = trans[i];

    float part = 0.0f;
    for (int tt = tid; tt < TLEN; tt += 256) {
        const float* lab = labels + ((size_t)b * TLEN + tt) * LDIM;
        const float* inp = inputs + ((size_t)b * TLEN + tt) * LDIM;
        float hs = 0.0f, fi = 0.0f;
        #pragma unroll
        for (int l = 0; l < LDIM; ++l) {
            const float lv = lab[l];
            hs = fmaf(inp[l], lv, hs);
            fi = fmaf((float)l, lv, fi);
        }
        idxs[tt] = (int)(fi + 0.5f);
        part += hs;
    }
    __syncthreads();

    for (int tt = tid; tt + 1 < TLEN; tt += 256)
        part += trs[idxs[tt] * LDIM + idxs[tt + 1]];

    red[tid] = part;
    __syncthreads();
    #pragma unroll
    for (int s = 128; s > 0; s >>= 1) {
        if (tid < s) red[tid] += red[tid + s];
        __syncthreads();
    }
    if (tid == 0) path[b] = red[0];
}

// ---------------------------------------------------------------------------
// Broadcast combine: out[i, j] = sum_over_path[j] - path_score[i]   (B x B)
// ---------------------------------------------------------------------------
__global__ __launch_bounds__(256) void crf_combine_kernel(
    const float* __restrict__ path,
    const float* __restrict__ sumpath,
    float* __restrict__ out)
{
    const int i = blockIdx.x * 256 + threadIdx.x;
    out[i] = sumpath[i & (BATCH - 1)] - path[i >> 10];
}

extern "C" void kernel_launch(void* const* d_in, const int* in_sizes, int n_in,
                              void* d_out, int out_size, void* d_ws, size_t ws_size,
                              hipStream_t stream)
{
    const float* inputs = (const float*)d_in[0];   // (B, T, L) f32
    const float* labels = (const float*)d_in[1];   // (B, T, L) f32 one-hot
    const float* trans  = (const float*)d_in[2];   // (L, L)    f32
    float* out = (float*)d_out;                    // (B, B)    f32

    float* path    = (float*)d_ws;                 // B floats
    float* sumpath = path + BATCH;                 // B floats

    crf_path_kernel<<<BATCH, 256, 0, stream>>>(inputs, labels, trans, path);
    crf_scan_kernel<<<BATCH / 16, 32, 0, stream>>>(inputs, trans, sumpath);
    crf_combine_kernel<<<(BATCH * BATCH) / 256, 256, 0, stream>>>(path, sumpath, out);
}